// MaskDICELoss_85237920956880
// MI455X (gfx1250) — compile-verified
//
#include <hip/hip_runtime.h>
#include <hip/hip_bf16.h>
#include <stdint.h>

#define LROW      65536
#define TILE      4096
#define NTHREADS  256
#define NTILES    (LROW / TILE)     // 16
#define EPT       (TILE / NTHREADS) // 16 elements per thread per tile
#define L2E       1.44269504088896f

typedef int v4i __attribute__((vector_size(16)));
typedef __attribute__((address_space(1))) v4i* g_v4i_ptr;
typedef __attribute__((address_space(3))) v4i* l_v4i_ptr;

// ---- CDNA5 async global -> LDS copy (16 bytes per lane), tracked by ASYNCcnt ----
__device__ __forceinline__ void async_copy16(const void* g, void* l) {
#if __has_builtin(__builtin_amdgcn_global_load_async_to_lds_b128)
    __builtin_amdgcn_global_load_async_to_lds_b128(
        (g_v4i_ptr)(uintptr_t)g,          // AS1: global/flat share addresses
        (l_v4i_ptr)(v4i*)l,               // addrspacecast generic -> AS3
        /*offset=*/0, /*cpol=*/0);
#else
    unsigned lds_off = (unsigned)(uintptr_t)(__attribute__((address_space(3))) char*)l;
    asm volatile("global_load_async_to_lds_b128 %0, %1, off"
                 :: "v"(lds_off), "v"(g) : "memory");
#endif
}

__device__ __forceinline__ void wait_async_le9() {
#if __has_builtin(__builtin_amdgcn_s_wait_asynccnt)
    __builtin_amdgcn_s_wait_asynccnt(9);
#else
    asm volatile("s_wait_asynccnt 9" ::: "memory");
#endif
}

__device__ __forceinline__ void wait_async_0() {
#if __has_builtin(__builtin_amdgcn_s_wait_asynccnt)
    __builtin_amdgcn_s_wait_asynccnt(0);
#else
    asm volatile("s_wait_asynccnt 0" ::: "memory");
#endif
}

// Online-softmax partial state and merge (rescale to common max).
struct OS { float m, s, set, st; };

__device__ __forceinline__ OS os_combine(OS a, OS b) {
    float M  = fmaxf(a.m, b.m);
    float ra = exp2f((a.m - M) * L2E);
    float rb = exp2f((b.m - M) * L2E);
    OS r;
    r.m   = M;
    r.s   = a.s   * ra + b.s   * rb;
    r.set = a.set * ra + b.set * rb;
    r.st  = a.st + b.st;
    return r;
}

__global__ __launch_bounds__(NTHREADS)
void dice_row_kernel(const float* __restrict__ logits,
                     const float* __restrict__ tmask,
                     const uint8_t* __restrict__ valid,
                     float* __restrict__ losses) {
    __shared__ __align__(16) float   lgs[2][TILE];
    __shared__ __align__(16) float   trs[2][TILE];
    __shared__ __align__(16) uint8_t vds[2][TILE];
    __shared__ float red[NTHREADS / 32][4];

    const int t = threadIdx.x;
    const int b = blockIdx.x;
    const size_t row = (size_t)b * LROW;

    const float*   gl = logits + row;
    const float*   gt = tmask  + row;
    const uint8_t* gv = valid  + row;

    // Stage one tile: each thread copies its own contiguous 64B+64B+16B slice.
    auto issue = [&](int tile, int buf) {
        const float*   pl = gl + tile * TILE + t * EPT;
        const float*   pt = gt + tile * TILE + t * EPT;
        const uint8_t* pv = gv + tile * TILE + t * EPT;
#pragma unroll
        for (int c = 0; c < 4; ++c) {
            async_copy16(pl + c * 4, &lgs[buf][t * EPT + c * 4]);
            async_copy16(pt + c * 4, &trs[buf][t * EPT + c * 4]);
        }
        async_copy16(pv, &vds[buf][t * EPT]);
    };

    issue(0, 0);
    issue(1, 1);

    float m = -1e30f, s = 0.0f, set = 0.0f, st = 0.0f;

    for (int n = 0; n < NTILES; ++n) {
        const int buf = n & 1;
        // Own 9 async ops of tile n are the oldest; ASYNCcnt loads retire in order.
        if (n < NTILES - 1) wait_async_le9();
        else                wait_async_0();
        __syncthreads();   // everyone's tile-n data now visible in LDS

        // Pass 1 (LDS -> registers): masked values, tile-local max, sum(t).
        float xs[EPT], ts[EPT];
        float tm = -1e30f;
#pragma unroll
        for (int c = 0; c < EPT; ++c) {
            int e = c * NTHREADS + t;          // lane-stride 4B: conflict-free ds loads
            float   x  = lgs[buf][e];
            float   tv = trs[buf][e];
            uint8_t v  = vds[buf][e];
            x  = v ? x  : -1e30f;
            tv = v ? tv : 0.0f;
            xs[c] = x; ts[c] = tv;
            tm = fmaxf(tm, x);
            st += tv;
        }
        // Rescale running sums once per tile, then 1 exp2 per element.
        float nm = fmaxf(m, tm);
        float r  = exp2f((m - nm) * L2E);
        s *= r; set *= r; m = nm;
#pragma unroll
        for (int c = 0; c < EPT; ++c) {
            float e = exp2f((xs[c] - m) * L2E);
            s   += e;
            set += e * ts[c];
        }

        __syncthreads();   // all done reading buf before it is overwritten
        if (n + 2 < NTILES) issue(n + 2, buf);
    }

    // wave32 reduction (xor shuffle tree), then cross-wave via LDS.
    OS a{m, s, set, st};
#pragma unroll
    for (int off = 16; off > 0; off >>= 1) {
        OS o;
        o.m   = __shfl_xor(a.m,   off, 32);
        o.s   = __shfl_xor(a.s,   off, 32);
        o.set = __shfl_xor(a.set, off, 32);
        o.st  = __shfl_xor(a.st,  off, 32);
        a = os_combine(a, o);
    }
    const int wave = t >> 5, lane = t & 31;
    if (lane == 0) {
        red[wave][0] = a.m; red[wave][1] = a.s;
        red[wave][2] = a.set; red[wave][3] = a.st;
    }
    __syncthreads();
    if (t == 0) {
        OS r0{red[0][0], red[0][1], red[0][2], red[0][3]};
        for (int w = 1; w < NTHREADS / 32; ++w) {
            OS q{red[w][0], red[w][1], red[w][2], red[w][3]};
            r0 = os_combine(r0, q);
        }
        // sum(p over valid) == 1 exactly (invalid mass underflows to 0 in f32)
        float num = 2.0f * r0.set / r0.s;
        float den = 1.0f + r0.st;
        losses[b] = 1.0f - (num + 1.0f) / (den + 1.0f);
    }
}

__global__ __launch_bounds__(512)
void mean_kernel(const float* __restrict__ losses, float* __restrict__ out, int B) {
    __shared__ float red[512];
    const int t = threadIdx.x;
    red[t] = (t < B) ? losses[t] : 0.0f;
    __syncthreads();
    for (int off = 256; off > 0; off >>= 1) {
        if (t < off) red[t] += red[t + off];
        __syncthreads();
    }
    if (t == 0) out[0] = red[0] / (float)B;
}

extern "C" void kernel_launch(void* const* d_in, const int* in_sizes, int n_in,
                              void* d_out, int out_size, void* d_ws, size_t ws_size,
                              hipStream_t stream) {
    const float*   logits = (const float*)d_in[0];
    const float*   tmask  = (const float*)d_in[1];
    const uint8_t* valid  = (const uint8_t*)d_in[2];   // jnp.bool_ -> 1 byte/elem
    const int B = in_sizes[0] / LROW;                  // 512

    float* losses = (float*)d_ws;                      // B floats of scratch

    dice_row_kernel<<<B, NTHREADS, 0, stream>>>(logits, tmask, valid, losses);
    mean_kernel<<<1, 512, 0, stream>>>(losses, (float*)d_out, B);
}